// SCCNNWrapper_49881750176305
// MI455X (gfx1250) — compile-verified
//
#include <hip/hip_runtime.h>

// ---------------------------------------------------------------------------
// SCCNN layer for MI455X (gfx1250, wave32).
// Dense GEMMs use V_WMMA_F32_16X16X4_F32 (full fp32, matches reference).
// GraphNorm collapsed to 2 passes via sum/sumsq identity (saves ~410MB pass).
// ---------------------------------------------------------------------------

typedef __attribute__((ext_vector_type(2))) float v2f;
typedef __attribute__((ext_vector_type(8))) float v8f;

#define CH        128      // feature channels
#define NGR       128      // number of graphs
#define GROWS     64       // GEMM rows per block (4 row-tiles of 16)
#define RT        4        // row tiles per wave
#define XSTRIDE   130      // padded LDS row stride (bank-conflict mitigation)
#define STAT_ROWS 64       // rows per stats block
#define EPS       1e-5f

// ---------------------------------------------------------------------------
// Y[nrows,128] = X[nrows,128] @ W[128,128]  via WMMA f32 16x16x4.
// Block = 256 threads = 8 waves. Block owns 64 rows; wave w owns cols
// [16w,16w+16) across 4 row tiles. B fragment reused across the 4 WMMAs
// per k-step; K=128 -> 128 WMMAs per wave.
// ---------------------------------------------------------------------------
__global__ __launch_bounds__(256)
void gemm_xw_wmma(const float* __restrict__ X, const float* __restrict__ W,
                  float* __restrict__ Y, int nrows)
{
    __shared__ float lW[CH * CH];           // 64 KB
    __shared__ float lX[GROWS * XSTRIDE];   // ~32.5 KB

    const int tid = threadIdx.x;
    for (int i = tid; i < CH * CH; i += 256) lW[i] = W[i];

    const int row0 = blockIdx.x * GROWS;
    for (int i = tid; i < GROWS * CH; i += 256) {
        int r = i >> 7, c = i & (CH - 1);
        int gr = row0 + r;
        lX[r * XSTRIDE + c] = (gr < nrows) ? X[(size_t)gr * CH + c] : 0.0f;
    }
    __syncthreads();

    const int wave = tid >> 5;
    const int lane = tid & 31;
    const int hi   = lane >> 4;    // 0: lanes 0-15, 1: lanes 16-31
    const int lo   = lane & 15;
    const int col0 = wave * 16;

    v8f acc[RT];
    #pragma unroll
    for (int t = 0; t < RT; ++t) acc[t] = (v8f){};

    #pragma unroll
    for (int kk = 0; kk < CH; kk += 4) {
        // B fragment (4x16): lane -> N=lo, VGPR v -> K = kk + v + 2*hi
        const int ka = kk + 2 * hi;
        v2f b;
        b.x = lW[(size_t)ka * CH + col0 + lo];
        b.y = lW[(size_t)(ka + 1) * CH + col0 + lo];
        #pragma unroll
        for (int t = 0; t < RT; ++t) {
            // A fragment (16x4): lane -> M=lo, VGPR v -> K = kk + v + 2*hi
            v2f a;
            a.x = lX[(t * 16 + lo) * XSTRIDE + ka];
            a.y = lX[(t * 16 + lo) * XSTRIDE + ka + 1];
            acc[t] = __builtin_amdgcn_wmma_f32_16x16x4_f32(
                         false, a, false, b, (short)0, acc[t], false, false);
        }
    }

    // D layout: VGPR r holds M = r + 8*hi, N = lo
    #pragma unroll
    for (int t = 0; t < RT; ++t) {
        #pragma unroll
        for (int r = 0; r < 8; ++r) {
            int row = row0 + t * 16 + r + 8 * hi;
            if (row < nrows) Y[(size_t)row * CH + col0 + lo] = acc[t][r];
        }
    }
}

// ---------------------------------------------------------------------------
// Grid-stride float4 copy.
// ---------------------------------------------------------------------------
__global__ void copy_f4(const float4* __restrict__ src, float4* __restrict__ dst,
                        size_t n4)
{
    size_t i = (size_t)blockIdx.x * blockDim.x + threadIdx.x;
    size_t stride = (size_t)gridDim.x * blockDim.x;
    for (; i < n4; i += stride) dst[i] = src[i];
}

// ---------------------------------------------------------------------------
// dst[rows[e], :] += vals[e] * src[cols[e], :]   (one wave per nnz entry;
// each lane handles 4 channels as a float4 gather + 4 atomic adds)
// ---------------------------------------------------------------------------
__global__ __launch_bounds__(256)
void spmm_scatter(const int* __restrict__ rows, const int* __restrict__ cols,
                  const float* __restrict__ vals, const float* __restrict__ src,
                  float* __restrict__ dst, int nnz)
{
    int w    = (int)(((size_t)blockIdx.x * blockDim.x + threadIdx.x) >> 5);
    int lane = threadIdx.x & 31;
    if (w >= nnz) return;

    int   r = rows[w];
    int   c = cols[w];
    float v = vals[w];

    const float4* s = (const float4*)(src + (size_t)c * CH);
    float*        d = dst + (size_t)r * CH + lane * 4;
    float4 t = s[lane];
    atomicAdd(d + 0, v * t.x);
    atomicAdd(d + 1, v * t.y);
    atomicAdd(d + 2, v * t.z);
    atomicAdd(d + 3, v * t.w);
}

// ---------------------------------------------------------------------------
// GraphNorm pass 1: per-graph channel sums, sums of squares, and counts.
// batch is sorted: block = 128 threads (one per channel) sweeps STAT_ROWS
// consecutive rows, flushing register accumulators to global atomics only
// at graph boundaries (~1-2 flushes per block).
// ---------------------------------------------------------------------------
__global__ __launch_bounds__(CH)
void gn_stats(const float* __restrict__ x, const int* __restrict__ batch,
              float* __restrict__ sum, float* __restrict__ sumsq,
              float* __restrict__ cnt, int nrows)
{
    const int ch = threadIdx.x;
    const int r0 = blockIdx.x * STAT_ROWS;
    const int re = min(r0 + STAT_ROWS, nrows);
    if (r0 >= nrows) return;

    int   g  = batch[r0];
    float s  = 0.0f;
    float s2 = 0.0f;
    int   c  = 0;
    for (int r = r0; r < re; ++r) {
        int gr = batch[r];
        if (gr != g) {
            atomicAdd(&sum[(size_t)g * CH + ch], s);
            atomicAdd(&sumsq[(size_t)g * CH + ch], s2);
            if (ch == 0) atomicAdd(&cnt[g], (float)c);
            s = 0.0f; s2 = 0.0f; c = 0; g = gr;
        }
        float v = x[(size_t)r * CH + ch];
        s  += v;
        s2 += v * v;
        ++c;
    }
    atomicAdd(&sum[(size_t)g * CH + ch], s);
    atomicAdd(&sumsq[(size_t)g * CH + ch], s2);
    if (ch == 0) atomicAdd(&cnt[g], (float)c);
}

// ---------------------------------------------------------------------------
// GraphNorm pass 2: in-place normalize.
//   a   = (sum/cnt) * mean_scale          (shifted mean)
//   var = E[x^2] - 2*a*E[x] + a^2  ==  E[(x - a)^2]   (reference variance)
//   out = weight * (x - a) / sqrt(var + eps) + bias
// ---------------------------------------------------------------------------
__global__ __launch_bounds__(256)
void gn_apply(float* __restrict__ x, const int* __restrict__ batch,
              const float* __restrict__ sum, const float* __restrict__ sumsq,
              const float* __restrict__ cnt,
              const float* __restrict__ weight, const float* __restrict__ bias,
              const float* __restrict__ mean_scale, int nrows)
{
    size_t idx = (size_t)blockIdx.x * blockDim.x + threadIdx.x;
    size_t total = (size_t)nrows * CH;
    if (idx >= total) return;
    int r  = (int)(idx >> 7);
    int ch = (int)(idx & (CH - 1));
    int g  = batch[r];
    float c   = fmaxf(cnt[g], 1.0f);
    float mu  = sum[(size_t)g * CH + ch] / c;
    float a   = mu * mean_scale[ch];
    float ex2 = sumsq[(size_t)g * CH + ch] / c;
    float var = fmaxf(ex2 - 2.0f * a * mu + a * a, 0.0f);
    float o   = x[idx] - a;
    float sd  = sqrtf(var + EPS);
    x[idx] = weight[ch] * o / sd + bias[ch];
}

// ---------------------------------------------------------------------------
// Host-side orchestration (graph-capture safe: only kernels + memsetAsync).
// ---------------------------------------------------------------------------
extern "C" void kernel_launch(void* const* d_in, const int* in_sizes, int n_in,
                              void* d_out, int out_size, void* d_ws, size_t ws_size,
                              hipStream_t stream)
{
    const float* x0       = (const float*)d_in[0];
    const float* x1       = (const float*)d_in[1];
    const float* x2       = (const float*)d_in[2];
    const float* W1       = (const float*)d_in[3];
    const float* W2       = (const float*)d_in[4];
    const float* gn1_w    = (const float*)d_in[5];
    const float* gn1_b    = (const float*)d_in[6];
    const float* gn1_ms   = (const float*)d_in[7];
    const float* gn2_w    = (const float*)d_in[8];
    const float* gn2_b    = (const float*)d_in[9];
    const float* gn2_ms   = (const float*)d_in[10];
    const int*   i2_rows  = (const int*)d_in[11];
    const int*   i2_cols  = (const int*)d_in[12];
    const float* i2_vals  = (const float*)d_in[13];
    const int*   i1_rows  = (const int*)d_in[14];
    const int*   i1_cols  = (const int*)d_in[15];
    const float* i1_vals  = (const float*)d_in[16];
    const int*   batch_n  = (const int*)d_in[17];
    const int*   batch_e  = (const int*)d_in[18];

    const int N0 = in_sizes[0] / CH;
    const int N1 = in_sizes[1] / CH;
    const int N2 = in_sizes[2] / CH;
    const int nnz2 = in_sizes[11];   // 3*N2
    const int nnz1 = in_sizes[14];   // 2*N1

    float* out    = (float*)d_out;
    float* y0     = out;                                  // x_0_out slot
    float* y1     = out + (size_t)N0 * CH;                // x_1_out slot
    float* out_x2 = out + (size_t)(N0 + N1) * CH;         // x_2 passthrough

    // Workspace: one shared GEMM result buffer + two stat sets
    // (cnt[NGR], sum[NGR*CH], sumsq[NGR*CH] each).
    size_t maxN   = (size_t)(N1 > N2 ? N1 : N2);
    float* xw     = (float*)d_ws;
    float* stats  = xw + maxN * CH;
    float* cnt1   = stats;
    float* sum1   = cnt1 + NGR;
    float* ssq1   = sum1 + (size_t)NGR * CH;
    float* cnt2   = ssq1 + (size_t)NGR * CH;
    float* sum2   = cnt2 + NGR;
    float* ssq2   = sum2 + (size_t)NGR * CH;
    size_t statBytes = sizeof(float) * (size_t)2 * (NGR + 2 * NGR * CH);
    hipMemsetAsync(stats, 0, statBytes, stream);

    // x_2 passthrough to output.
    copy_f4<<<4096, 256, 0, stream>>>((const float4*)x2, (float4*)out_x2,
                                      (size_t)N2 * CH / 4);

    // ---- hop 1: edges <- triangles --------------------------------------
    gemm_xw_wmma<<<(N2 + GROWS - 1) / GROWS, 256, 0, stream>>>(x2, W1, xw, N2);
    copy_f4<<<4096, 256, 0, stream>>>((const float4*)x1, (float4*)y1,
                                      (size_t)N1 * CH / 4);
    spmm_scatter<<<((size_t)nnz2 * 32 + 255) / 256, 256, 0, stream>>>(
        i2_rows, i2_cols, i2_vals, xw, y1, nnz2);

    gn_stats<<<(N1 + STAT_ROWS - 1) / STAT_ROWS, CH, 0, stream>>>(
        y1, batch_e, sum1, ssq1, cnt1, N1);
    gn_apply<<<((size_t)N1 * CH + 255) / 256, 256, 0, stream>>>(
        y1, batch_e, sum1, ssq1, cnt1, gn1_w, gn1_b, gn1_ms, N1);

    // ---- hop 2: nodes <- edges ------------------------------------------
    gemm_xw_wmma<<<(N1 + GROWS - 1) / GROWS, 256, 0, stream>>>(y1, W2, xw, N1);
    copy_f4<<<4096, 256, 0, stream>>>((const float4*)x0, (float4*)y0,
                                      (size_t)N0 * CH / 4);
    spmm_scatter<<<((size_t)nnz1 * 32 + 255) / 256, 256, 0, stream>>>(
        i1_rows, i1_cols, i1_vals, xw, y0, nnz1);

    gn_stats<<<(N0 + STAT_ROWS - 1) / STAT_ROWS, CH, 0, stream>>>(
        y0, batch_n, sum2, ssq2, cnt2, N0);
    gn_apply<<<((size_t)N0 * CH + 255) / 256, 256, 0, stream>>>(
        y0, batch_n, sum2, ssq2, cnt2, gn2_w, gn2_b, gn2_ms, N0);
}